// ConduitHydrology_15814069584535
// MI455X (gfx1250) — compile-verified
//
#include <hip/hip_runtime.h>
#include <hip/hip_bf16.h>

// ConduitHydrology on MI455X (gfx1250, wave32).
// Fixed-iteration (100) device-side CG for the 5-point Laplacian on a
// 2048x2048 grid. All CG state (r, p double-buffered, Ap) lives in d_ws and
// fits in the 192MB L2, so the solve is L2-BW bound. Dot products use
// V_WMMA_F32_16X16X4_F32 (ones-matrix trick) for the wave32 reduction.

typedef __attribute__((ext_vector_type(2))) float v2f;
typedef __attribute__((ext_vector_type(8))) float v8f;

#define NRr 2048
#define NCc 2048
#define NTOT (NRr * NCc)
#define CG_ITERS 100

static constexpr float FLOW_COEFF = 0.0405f;
static constexpr float DXC        = 100.0f;
static constexpr float INV_DX     = 1.0f / 100.0f;
static constexpr float INV_DX2    = 1.0f / (100.0f * 100.0f);

// ---------------------------------------------------------------------------
// Wave32 sum via V_WMMA_F32_16X16X4_F32:
// A (16x4 f32): lanes 0-15 hold {K0,K1} = {s,0}; lanes 16-31 hold {K2,K3}={s,0}
//   -> row m of A = [s_m, 0, s_{16+m}, 0]
// B = ones(4x16)  -> D[m][n] = s_m + s_{16+m} for every n.
// D layout: lane n, VGPR v = D[v + (lane>=16 ? 8:0)][n]; summing the 8 C regs
// per lane then adding across the half-waves (xor 16) gives the full wave sum.
// EXEC must be all ones at this point (callers guarantee no divergence).
// ---------------------------------------------------------------------------
__device__ __forceinline__ float wave_sum_wmma(float s) {
  v2f a;    a[0] = s;    a[1] = 0.0f;
  v2f ones; ones[0] = 1.0f; ones[1] = 1.0f;
  v8f c = {};
  c = __builtin_amdgcn_wmma_f32_16x16x4_f32(
      /*neg_a=*/false, a, /*neg_b=*/false, ones,
      /*c_mod=*/(short)0, c, /*reuse_a=*/false, /*reuse_b=*/false);
  float t = ((c[0] + c[1]) + (c[2] + c[3])) + ((c[4] + c[5]) + (c[6] + c[7]));
  t += __shfl_xor(t, 16, 32);   // combine lane halves -> full wave32 sum
  return t;
}

// Block = 256 threads = 8 wave32s.
__device__ __forceinline__ void block_reduce_atomic(float v, float* target) {
  __shared__ float smem[8];
  float w = wave_sum_wmma(v);
  int lane = threadIdx.x & 31;
  int wid  = threadIdx.x >> 5;
  if (lane == 0) smem[wid] = w;
  __syncthreads();
  if (threadIdx.x == 0) {
    float t = 0.0f;
#pragma unroll
    for (int i = 0; i < 8; ++i) t += smem[i];
    atomicAdd(target, t);
  }
}

__device__ __forceinline__ bool is_bnd(int rr, int cc) {
  return (rr == 0) | (rr == NRr - 1) | (cc == 0) | (cc == NCc - 1);
}

// grad_nodes = (q * 0.0405 * cs^1.25)^2 ; cs^1.25 = cs * sqrt(sqrt(cs))
__device__ __forceinline__ float node_G(const float* cs, const float* q, int idx) {
  float c = cs[idx];
  float t = q[idx] * FLOW_COEFF * (c * __builtin_sqrtf(__builtin_sqrtf(c)));
  return t * t;
}

__device__ __forceinline__ float link_g(const float* cs, const float* q,
                                        const float* gg, int ia, int ib,
                                        bool bnd_any) {
  if (bnd_any) return 0.5f * (gg[ia] + gg[ib]);
  return 0.5f * (node_G(cs, q, ia) + node_G(cs, q, ib));
}

// ---------------------------------------------------------------------------

__global__ void k_zero_scalars(float* scal) {
  int i = threadIdx.x;
  if (i < 2 * CG_ITERS + 2) scal[i] = 0.0f;
}

// b = div_f ; x = 0 ; r = p = b ; scal[0] += sum(b*b)
__global__ void k_init(const float* __restrict__ cs, const float* __restrict__ q,
                       const float* __restrict__ gg, float* __restrict__ x,
                       float* __restrict__ r_arr, float* __restrict__ p_arr,
                       float* scal) {
  int n  = blockIdx.x * blockDim.x + threadIdx.x;
  int rr = n >> 11, cc = n & (NCc - 1);
  bool b0 = is_bnd(rr, cc);
  float acc = 0.0f;
  if (cc < NCc - 1) acc += link_g(cs, q, gg, n, n + 1,   b0 || is_bnd(rr, cc + 1)); // tail of right link
  if (rr < NRr - 1) acc += link_g(cs, q, gg, n, n + NCc, b0 || is_bnd(rr + 1, cc)); // tail of up link
  if (cc > 0)       acc -= link_g(cs, q, gg, n - 1,   n, b0 || is_bnd(rr, cc - 1)); // head of left link
  if (rr > 0)       acc -= link_g(cs, q, gg, n - NCc, n, b0 || is_bnd(rr - 1, cc)); // head of down link
  float b = acc * INV_DX;
  x[n] = 0.0f;
  r_arr[n] = b;
  p_arr[n] = b;
  block_reduce_atomic(b * b, &scal[0]);   // rr_0
}

// p_new = r + beta*p_old (beta = rr_i / rr_{i-1}, 0 for i==0);
// Ap = L(p_new) with neighbor p_new recomputed on the fly (race-free via
// double buffer); scal[2i+1] += sum(p_new * Ap).
__global__ void k_fusedA(const float* __restrict__ r_arr,
                         const float* __restrict__ p_old,
                         float* __restrict__ p_new, float* __restrict__ Ap,
                         float* scal, int iter) {
  int n  = blockIdx.x * blockDim.x + threadIdx.x;
  int rr = n >> 11, cc = n & (NCc - 1);
  float beta = 0.0f;
  if (iter > 0) beta = scal[2 * iter] / scal[2 * iter - 2];
  float center = r_arr[n] + beta * p_old[n];
  float lap = 0.0f;
  if (cc < NCc - 1) lap += (r_arr[n + 1]   + beta * p_old[n + 1])   - center;
  if (cc > 0)       lap += (r_arr[n - 1]   + beta * p_old[n - 1])   - center;
  if (rr < NRr - 1) lap += (r_arr[n + NCc] + beta * p_old[n + NCc]) - center;
  if (rr > 0)       lap += (r_arr[n - NCc] + beta * p_old[n - NCc]) - center;
  float ap = lap * INV_DX2;
  p_new[n] = center;
  Ap[n]    = ap;
  block_reduce_atomic(center * ap, &scal[2 * iter + 1]);  // pAp_i
}

// alpha = rr_i / pAp_i ; x += alpha*p ; r -= alpha*Ap ; scal[2i+2] += sum(r*r)
__global__ void k_fusedB(const float* __restrict__ p_new,
                         const float* __restrict__ Ap, float* __restrict__ x,
                         float* __restrict__ r_arr, float* scal, int iter) {
  int n = blockIdx.x * blockDim.x + threadIdx.x;
  float alpha = scal[2 * iter] / scal[2 * iter + 1];
  float rn = r_arr[n] - alpha * Ap[n];
  x[n] += alpha * p_new[n];
  r_arr[n] = rn;
  block_reduce_atomic(rn * rn, &scal[2 * iter + 2]);      // rr_{i+1}
}

// out = geometric_gradient - potential (in place: x lives in d_out)
__global__ void k_final(const float* __restrict__ gg, float* __restrict__ x) {
  int n = blockIdx.x * blockDim.x + threadIdx.x;
  x[n] = gg[n] - x[n];
}

// ---------------------------------------------------------------------------

extern "C" void kernel_launch(void* const* d_in, const int* in_sizes, int n_in,
                              void* d_out, int out_size, void* d_ws, size_t ws_size,
                              hipStream_t stream) {
  const float* cs = (const float*)d_in[0];  // conduit_size
  const float* q  = (const float*)d_in[1];  // discharge
  const float* gg = (const float*)d_in[2];  // geometric_gradient
  // d_in[3]/d_in[4] = nrows/ncols (2048/2048, hardcoded for shift addressing)

  float* x    = (float*)d_out;              // potential accumulates here
  float* wsf  = (float*)d_ws;
  float* r    = wsf;                        // residual            (N floats)
  float* pA   = wsf + (size_t)NTOT;         // search dir buffer A (N floats)
  float* pB   = wsf + 2 * (size_t)NTOT;     // search dir buffer B (N floats)
  float* Ap   = wsf + 3 * (size_t)NTOT;     // L(p)                (N floats)
  float* scal = wsf + 4 * (size_t)NTOT;     // [rr_0, pAp_0, rr_1, pAp_1, ...]

  dim3 blk(256);
  dim3 grd(NTOT / 256);   // exact: 16384 blocks, no bounds divergence (WMMA EXEC)

  k_zero_scalars<<<dim3(1), blk, 0, stream>>>(scal);
  k_init<<<grd, blk, 0, stream>>>(cs, q, gg, x, r, pA, scal);

  for (int i = 0; i < CG_ITERS; ++i) {
    const float* po = (i & 1) ? pB : pA;
    float*       pn = (i & 1) ? pA : pB;
    k_fusedA<<<grd, blk, 0, stream>>>(r, po, pn, Ap, scal, i);
    k_fusedB<<<grd, blk, 0, stream>>>(pn, Ap, x, r, scal, i);
  }

  k_final<<<grd, blk, 0, stream>>>(gg, x);
}